// TttM2BMMTritonModule_61701500174859
// MI455X (gfx1250) — compile-verified
//
#include <hip/hip_runtime.h>
#include <stdint.h>

// TTT-M2 sequential decode scan, MI455X (gfx1250, wave32).
// One workgroup (256 threads, 8 waves) per (b,h) head; 256 workgroups.
// W1/W1g column-per-thread + W2g row-per-thread in VGPRs; W2 in LDS with
// stride-65 padded rows (bank-conflict-free row AND column access).
// CDNA5 paths: TDM tensor_load_to_lds (pad_enable creates the stride-65
// layout in-flight), global_load_async_to_lds_* double-buffering of the
// per-token xa/xb/xc/coeff, s_wait_asynccnt/s_wait_tensorcnt.
// 5 barriers per token (A..E); the async wait is folded into barrier E.

static constexpr int kB   = 8;
static constexpr int kNH  = 32;
static constexpr int kL   = 512;
static constexpr int kHF  = 64;
static constexpr int kHFP = 256;
static constexpr int kW2S = 65;   // padded LDS row stride (floats)

typedef uint32_t v4u __attribute__((ext_vector_type(4)));
typedef int      v8i __attribute__((ext_vector_type(8)));
typedef int      v4i __attribute__((ext_vector_type(4)));

struct Lds {
  float W2s[kHFP * kW2S];   // 66560 B, stride-65 rows
  float Z1s[kHFP];
  float Z1bs[kHFP];
  float gZ2s[kHF];
  float red[4 * kHF];
  float xbuf[2][3][kHF];    // [buf][0=xa,1=xb,2=xc]
  float cbuf[2];
};

__device__ __forceinline__ uint32_t lds_off(const void* p) {
  return (uint32_t)(uintptr_t)p;   // low 32 bits of flat shared addr = LDS byte offset
}

__device__ __forceinline__ void async_b128(uint32_t ldst, uint32_t goff, const void* sbase) {
  asm volatile("global_load_async_to_lds_b128 %0, %1, %2"
               :: "v"(ldst), "v"(goff), "s"(sbase) : "memory");
}
__device__ __forceinline__ void async_b32(uint32_t ldst, uint32_t goff, const void* sbase) {
  asm volatile("global_load_async_to_lds_b32 %0, %1, %2"
               :: "v"(ldst), "v"(goff), "s"(sbase) : "memory");
}
__device__ __forceinline__ void wait_async0() {
  asm volatile("s_wait_asynccnt 0x0" ::: "memory");
}

// TDM: copy W2[256][64] (fp32, contiguous) -> LDS with 1-DWORD pad after each
// 64-DWORD row, producing the stride-65 layout directly.
__device__ __forceinline__ void tdm_load_w2(const float* gsrc, void* ldst) {
  uint64_t ga = (uint64_t)(uintptr_t)gsrc;
  v4u g0;
  g0[0] = 1u;                                             // count=1 (valid), user mode
  g0[1] = lds_off(ldst);                                  // lds_addr (bytes)
  g0[2] = (uint32_t)ga;                                   // global_addr[31:0]
  g0[3] = (uint32_t)((ga >> 32) & 0x1FFFFFFull) | (2u << 30); // addr[56:32] | type=2
  v8i g1;
  g1[0] = (int)((2u << 16)        // data_size = 4 bytes
              | (1u << 20)        // pad_enable
              | (5u << 22)        // pad_interval: 64 DWORDs
              | (0u << 25));      // pad_amount: 1 DWORD
  g1[1] = (int)(64u  << 16);      // tensor_dim0 = 64  (bits 79:48, low half)
  g1[2] = (int)(256u << 16);      // tensor_dim1 = 256 (bits 111:80, low half)
  g1[3] = (int)(64u  << 16);      // tile_dim0 = 64 (bits 127:112)
  g1[4] = (int)256;               // tile_dim1 = 256, tile_dim2 = 0
  g1[5] = (int)64;                // tensor_dim0_stride = 64 elements
  g1[6] = (int)(16384u << 16);    // tensor_dim1_stride = 64*256 (low 16)
  g1[7] = 0;                      // tensor_dim1_stride hi = 0
  v4i gz4 = {0, 0, 0, 0};
  v8i gz8 = {0, 0, 0, 0, 0, 0, 0, 0};
  // clang-23 / therock-10.0 toolchain: 6-arg form
  // (uint32x4 g0, int32x8 g1, int32x4, int32x4, int32x8, i32 cpol)
  __builtin_amdgcn_tensor_load_to_lds(g0, g1, gz4, gz4, gz8, 0);
}

__device__ __forceinline__ void issue_token_loads(
    int t, int bh, int tok, int buf, Lds* lds,
    const float* XA, const float* XB, const float* XC, const float* coeff) {
  const int w = t >> 5, l = t & 31;
  if (w < 3) {
    if (l < 16) {
      uint32_t goff = (uint32_t)(((((uint32_t)bh * kL + tok) * kHF) + l * 4) * 4u);
      uint32_t ldst = lds_off(&lds->xbuf[buf][w][l * 4]);
      if (w == 0)      async_b128(ldst, goff, XA);
      else if (w == 1) async_b128(ldst, goff, XB);
      else             async_b128(ldst, goff, XC);
    }
  } else if (t == 96) {
    uint32_t goff = (uint32_t)(((uint32_t)bh * kL + tok) * 4u);
    async_b32(lds_off(&lds->cbuf[buf]), goff, coeff);
  }
}

__global__ __launch_bounds__(256)
void ttt_m2_scan_kernel(const float* __restrict__ XA, const float* __restrict__ XB,
                        const float* __restrict__ XC, const float* __restrict__ coeff,
                        const float* __restrict__ W1, const float* __restrict__ W2,
                        const float* __restrict__ W1g, const float* __restrict__ W2g,
                        float* __restrict__ Out) {
  __shared__ Lds lds;
  const int t  = threadIdx.x;
  const int bh = blockIdx.x;
  const int j  = t & 63;
  const int k  = t >> 6;

  // ---- register-resident state ----
  float w1c[kHF], w1gc[kHF], w2gr[kHF];
  {
    const float* W1p  = W1  + (size_t)bh * kHF * kHFP;
    const float* W1gp = W1g + (size_t)bh * kHF * kHFP;
#pragma unroll
    for (int f = 0; f < kHF; ++f) {
      w1c[f]  = W1p [f * kHFP + t];   // column t of W1
      w1gc[f] = W1gp[f * kHFP + t];   // column t of W1_grad
    }
    const float* W2gp = W2g + ((size_t)bh * kHFP + t) * kHF;
#pragma unroll
    for (int q = 0; q < kHF; ++q) w2gr[q] = W2gp[q];  // row t of W2_grad
  }

  // ---- TDM load of W2 tile into padded LDS (issued by wave 0 only) ----
  if (t < 32) tdm_load_w2(W2 + (size_t)bh * kHFP * kHF, lds.W2s);
  __builtin_amdgcn_s_wait_tensorcnt(0);

  // ---- preload token 0 inputs (async) ----
  issue_token_loads(t, bh, 0, 0, &lds, XA, XB, XC, coeff);
  wait_async0();
  __syncthreads();

  float* myrow = &lds.W2s[t * kW2S];
  const float* w2col = &lds.W2s[(k << 6) * kW2S + j];

  for (int tok = 0; tok < kL; ++tok) {
    const int cur = tok & 1;
    // Issue next token's DMA now; it overlaps phases (1)..(7) and is waited
    // just before barrier E. Safe: all reads of xbuf[cur^1] ended before the
    // previous iteration's barrier D < E < this issue.
    if (tok + 1 < kL)
      issue_token_loads(t, bh, tok + 1, cur ^ 1, &lds, XA, XB, XC, coeff);

    const float* xa = lds.xbuf[cur][0];
    const float* xb = lds.xbuf[cur][1];
    const float* xc = lds.xbuf[cur][2];

    // (1) Z1[t] = sum_f xb[f] * W1[f][t]        (registers + LDS broadcast)
    float a0 = 0.f, a1 = 0.f, a2 = 0.f, a3 = 0.f;
#pragma unroll
    for (int f = 0; f < kHF; f += 4) {
      a0 = fmaf(xb[f + 0], w1c[f + 0], a0);
      a1 = fmaf(xb[f + 1], w1c[f + 1], a1);
      a2 = fmaf(xb[f + 2], w1c[f + 2], a2);
      a3 = fmaf(xb[f + 3], w1c[f + 3], a3);
    }
    const float z1_t = (a0 + a1) + (a2 + a3);
    lds.Z1s[t] = z1_t;
    __syncthreads();  // A: Z1 complete

    // (2) Z2 partials: thread t -> output j, quarter k of the 256-long sum
    {
      const float* z1k = &lds.Z1s[k << 6];
      float p0 = 0.f, p1 = 0.f, p2 = 0.f, p3 = 0.f;
#pragma unroll
      for (int i = 0; i < 64; i += 4) {
        p0 = fmaf(z1k[i + 0], w2col[(i + 0) * kW2S], p0);
        p1 = fmaf(z1k[i + 1], w2col[(i + 1) * kW2S], p1);
        p2 = fmaf(z1k[i + 2], w2col[(i + 2) * kW2S], p2);
        p3 = fmaf(z1k[i + 3], w2col[(i + 3) * kW2S], p3);
      }
      lds.red[t] = (p0 + p1) + (p2 + p3);
    }
    __syncthreads();  // B: partials complete

    // (3) reduce Z2, form gZ2 = Z2 - xa
    if (t < kHF) {
      float z2 = (lds.red[t] + lds.red[64 + t]) + (lds.red[128 + t] + lds.red[192 + t]);
      lds.gZ2s[t] = z2 - xa[t];
    }
    __syncthreads();  // C: gZ2 complete

    const float c = lds.cbuf[cur];

    // (4) gZ1[t] = sum_j gZ2[j] * W2[t][j]      (own padded LDS row)
    float g0 = 0.f, g1 = 0.f, g2 = 0.f, g3 = 0.f;
#pragma unroll
    for (int q = 0; q < kHF; q += 4) {
      g0 = fmaf(lds.gZ2s[q + 0], myrow[q + 0], g0);
      g1 = fmaf(lds.gZ2s[q + 1], myrow[q + 1], g1);
      g2 = fmaf(lds.gZ2s[q + 2], myrow[q + 2], g2);
      g3 = fmaf(lds.gZ2s[q + 3], myrow[q + 3], g3);
    }
    const float gz1_t = (g0 + g1) + (g2 + g3);

    // (5) W1g += xb (x) gZ1 ; W1 -= c*W1g ; Z1b[t] = sum_f xc[f]*W1new[f][t]
    float b0 = 0.f, b1 = 0.f, b2 = 0.f, b3 = 0.f;
#pragma unroll
    for (int f = 0; f < kHF; f += 4) {
      w1gc[f + 0] = fmaf(xb[f + 0], gz1_t, w1gc[f + 0]);
      w1c [f + 0] = fmaf(-c, w1gc[f + 0], w1c[f + 0]);
      b0 = fmaf(xc[f + 0], w1c[f + 0], b0);
      w1gc[f + 1] = fmaf(xb[f + 1], gz1_t, w1gc[f + 1]);
      w1c [f + 1] = fmaf(-c, w1gc[f + 1], w1c[f + 1]);
      b1 = fmaf(xc[f + 1], w1c[f + 1], b1);
      w1gc[f + 2] = fmaf(xb[f + 2], gz1_t, w1gc[f + 2]);
      w1c [f + 2] = fmaf(-c, w1gc[f + 2], w1c[f + 2]);
      b2 = fmaf(xc[f + 2], w1c[f + 2], b2);
      w1gc[f + 3] = fmaf(xb[f + 3], gz1_t, w1gc[f + 3]);
      w1c [f + 3] = fmaf(-c, w1gc[f + 3], w1c[f + 3]);
      b3 = fmaf(xc[f + 3], w1c[f + 3], b3);
    }
    lds.Z1bs[t] = (b0 + b1) + (b2 + b3);

    // (6) W2g row += Z1[t] * gZ2 ; W2 row -= c * W2g row   (own row only)
#pragma unroll
    for (int q = 0; q < kHF; ++q) {
      const float gq = lds.gZ2s[q];
      w2gr[q] = fmaf(z1_t, gq, w2gr[q]);
      myrow[q] = fmaf(-c, w2gr[q], myrow[q]);
    }
    __syncthreads();  // D: Z1b + W2 update complete

    // (7) out partials: out[j] = sum_p Z1b[p] * W2new[p][j]
    {
      const float* zbk = &lds.Z1bs[k << 6];
      float o0 = 0.f, o1 = 0.f, o2 = 0.f, o3 = 0.f;
#pragma unroll
      for (int i = 0; i < 64; i += 4) {
        o0 = fmaf(zbk[i + 0], w2col[(i + 0) * kW2S], o0);
        o1 = fmaf(zbk[i + 1], w2col[(i + 1) * kW2S], o1);
        o2 = fmaf(zbk[i + 2], w2col[(i + 2) * kW2S], o2);
        o3 = fmaf(zbk[i + 3], w2col[(i + 3) * kW2S], o3);
      }
      lds.red[t] = (o0 + o1) + (o2 + o3);
    }
    wait_async0();    // next-token DMA landed; barrier E publishes it to all
    __syncthreads();  // E: out partials + next-token inputs visible

    // (8) reduce + store output. red's next writer is behind barrier A of the
    // next iteration, and all xbuf[cur] readers finished before D, so no
    // trailing barrier is needed here.
    if (t < kHF) {
      float r = (lds.red[t] + lds.red[64 + t]) + (lds.red[128 + t] + lds.red[192 + t]);
      Out[((size_t)bh * kL + tok) * kHF + t] = r;
    }
  }
}

extern "C" void kernel_launch(void* const* d_in, const int* in_sizes, int n_in,
                              void* d_out, int out_size, void* d_ws, size_t ws_size,
                              hipStream_t stream) {
  (void)in_sizes; (void)n_in; (void)out_size; (void)d_ws; (void)ws_size;
  const float* XA    = (const float*)d_in[0];
  const float* XB    = (const float*)d_in[1];
  const float* XC    = (const float*)d_in[2];
  const float* coeff = (const float*)d_in[3];
  const float* W1    = (const float*)d_in[4];
  const float* W2    = (const float*)d_in[5];
  const float* W1g   = (const float*)d_in[6];
  const float* W2g   = (const float*)d_in[7];
  dim3 grid(kB * kNH), block(256);
  hipLaunchKernelGGL(ttt_m2_scan_kernel, grid, block, 0, stream,
                     XA, XB, XC, coeff, W1, W2, W1g, W2g, (float*)d_out);
}